// DConv_15685220565128
// MI455X (gfx1250) — compile-verified
//
#include <hip/hip_runtime.h>

#define NN   16384       // nodes
#define EE   524288      // edges
#define BB   32          // batch
#define DD   64          // input dim
#define OUTD 64          // output dim
#define MMAT 5           // num matrices (2K+1)
#define FF   2048        // D*B features per node
#define CH   ((size_t)NN * FF)   // elems per xs plane

typedef __attribute__((ext_vector_type(2))) float v2f;
typedef __attribute__((ext_vector_type(8))) float v8f;

// ---- degrees (float atomics) + per-destination edge histograms (int atomics) ----
__global__ __launch_bounds__(256) void k_deg_cnt(const float* __restrict__ adj,
                                                 const int* __restrict__ rows,
                                                 const int* __restrict__ cols,
                                                 float* __restrict__ drow,
                                                 float* __restrict__ dcol,
                                                 int* __restrict__ cnt1,   // keyed by cols (S1 dst)
                                                 int* __restrict__ cnt2) { // keyed by rows (S2 dst)
    int e = blockIdx.x * 256 + threadIdx.x;
    if (e < EE) {
        float a = adj[e];
        int r = rows[e], c = cols[e];
        unsafeAtomicAdd(&drow[r], a);
        unsafeAtomicAdd(&dcol[c], a);
        atomicAdd(&cnt1[c], 1);
        atomicAdd(&cnt2[r], 1);
    }
}

// ---- exclusive scan of 16384 counts, single block; blockIdx picks which support ----
__global__ __launch_bounds__(256) void k_scan(const int* __restrict__ cnt1, int* __restrict__ off1, int* __restrict__ cur1,
                                              const int* __restrict__ cnt2, int* __restrict__ off2, int* __restrict__ cur2) {
    const int* cnt = blockIdx.x ? cnt2 : cnt1;
    int* off = blockIdx.x ? off2 : off1;
    int* cur = blockIdx.x ? cur2 : cur1;
    __shared__ int part[256];
    const int C = NN / 256;   // 64 per thread
    int t = threadIdx.x;
    int s = 0;
    for (int i = 0; i < C; ++i) s += cnt[t * C + i];
    part[t] = s;
    __syncthreads();
    if (t == 0) {
        int run = 0;
        for (int i = 0; i < 256; ++i) { int tmp = part[i]; part[i] = run; run += tmp; }
    }
    __syncthreads();
    int run = part[t];
    for (int i = 0; i < C; ++i) {
        int idx = t * C + i;
        off[idx] = run; cur[idx] = run;
        run += cnt[idx];
    }
    if (t == 0) off[NN] = EE;
}

// ---- fill packed (src, normalized val) buckets for both supports ----
__global__ __launch_bounds__(256) void k_fill(const float* __restrict__ adj,
                                              const int* __restrict__ rows,
                                              const int* __restrict__ cols,
                                              const float* __restrict__ drow,
                                              const float* __restrict__ dcol,
                                              int* __restrict__ cur1, int* __restrict__ src1p, float* __restrict__ val1p,
                                              int* __restrict__ cur2, int* __restrict__ src2p, float* __restrict__ val2p) {
    int e = blockIdx.x * 256 + threadIdx.x;
    if (e < EE) {
        float a = adj[e];
        int r = rows[e], c = cols[e];
        float dr = drow[r], dc = dcol[c];
        float v1 = a * (dr > 0.f ? 1.f / dr : 0.f);   // S1: gather rows, scatter cols
        float v2 = a * (dc > 0.f ? 1.f / dc : 0.f);   // S2: gather cols, scatter rows
        int p1 = atomicAdd(&cur1[c], 1);
        src1p[p1] = r; val1p[p1] = v1;
        int p2 = atomicAdd(&cur2[r], 1);
        src2p[p2] = c; val2p[p2] = v2;
    }
}

// ---- pack x0: [B,N,D] -> [N, D*B] via padded LDS transpose ----
__global__ __launch_bounds__(256) void k_pack_x0(const float* __restrict__ in,
                                                 float* __restrict__ x0) {
    __shared__ float tile[DD * 33];
    int n = blockIdx.x;
    for (int r = threadIdx.x; r < FF; r += 256) {
        int b = r >> 6, d = r & 63;     // d fastest -> 256B coalesced reads
        tile[d * 33 + b] = in[(size_t)b * (NN * DD) + (size_t)n * DD + d];
    }
    __syncthreads();
    size_t base = (size_t)n * FF;
    for (int f = threadIdx.x; f < FF; f += 256) {
        int d = f >> 5, b = f & 31;
        x0[base + f] = tile[d * 33 + b];
    }
}

// ---- pull-mode SpMM: y[n] = scale * sum_{p in bucket(n)} val[p]*x[src[p]]  (- xprev[n]) ----
// One block per destination node; 256 threads x 2 float4 accumulators cover F=2048.
// No float atomics: plain coalesced b128 gathers + b128 stores.
__global__ __launch_bounds__(256) void k_spmm_gather(const float* __restrict__ x,
                                                     const float* __restrict__ xprev,
                                                     float* __restrict__ y,
                                                     const int* __restrict__ off,
                                                     const int* __restrict__ srcp,
                                                     const float* __restrict__ valp,
                                                     float scale, int useprev) {
    int n = blockIdx.x, t = threadIdx.x;
    int p = off[n], pe = off[n + 1];
    float4 a0 = make_float4(0.f, 0.f, 0.f, 0.f);
    float4 a1 = make_float4(0.f, 0.f, 0.f, 0.f);
    for (; p < pe; ++p) {
        int s = srcp[p];
        float v = valp[p];
        const float4* xp = (const float4*)(x + (size_t)s * FF);
        float4 u0 = xp[t], u1 = xp[256 + t];
        a0.x += v * u0.x; a0.y += v * u0.y; a0.z += v * u0.z; a0.w += v * u0.w;
        a1.x += v * u1.x; a1.y += v * u1.y; a1.z += v * u1.z; a1.w += v * u1.w;
    }
    a0.x *= scale; a0.y *= scale; a0.z *= scale; a0.w *= scale;
    a1.x *= scale; a1.y *= scale; a1.z *= scale; a1.w *= scale;
    if (useprev) {   // Chebyshev combine: y = scale*Ax - xprev
        const float4* pp = (const float4*)(xprev + (size_t)n * FF);
        float4 q0 = pp[t], q1 = pp[256 + t];
        a0.x -= q0.x; a0.y -= q0.y; a0.z -= q0.z; a0.w -= q0.w;
        a1.x -= q1.x; a1.y -= q1.y; a1.z -= q1.z; a1.w -= q1.w;
    }
    float4* yp = (float4*)(y + (size_t)n * FF);
    yp[t] = a0;
    yp[256 + t] = a1;
}

// ---- WMMA f32 GEMM: out[b*N+n, o] = sum_k X[.,k]*W[k,o] + bias ----
// K permuted to k' = m*64 + d on both A and W (product invariant) -> coalesced A loads.
__global__ __launch_bounds__(256) void k_gemm_wmma(const float* __restrict__ xs,
                                                   const float* __restrict__ W,
                                                   const float* __restrict__ bias,
                                                   float* __restrict__ out) {
    int n    = blockIdx.x;
    int wid  = threadIdx.x >> 5;
    int lane = threadIdx.x & 31;
    int bh = wid >> 2;            // batch half: b 0..15 / 16..31
    int ot = wid & 3;             // 16-wide output column tile
    int l  = lane & 15;
    int hi = lane >> 4;           // lane half carries K 0/1 vs 2/3 of each group
    int k0 = hi * 2;
    int bbase = bh * 16;
    int obase = ot * 16;

    v8f c = {0.f, 0.f, 0.f, 0.f, 0.f, 0.f, 0.f, 0.f};

    for (int m = 0; m < MMAT; ++m) {
        const float* xm = xs + (size_t)m * CH + (size_t)n * FF + bbase + l;
        #pragma unroll
        for (int d0 = 0; d0 < DD; d0 += 4) {
            v2f a, b;
            a.x = xm[(d0 + k0    ) * BB];
            a.y = xm[(d0 + k0 + 1) * BB];
            b.x = W[((d0 + k0    ) * MMAT + m) * OUTD + obase + l];
            b.y = W[((d0 + k0 + 1) * MMAT + m) * OUTD + obase + l];
            c = __builtin_amdgcn_wmma_f32_16x16x4_f32(
                    false, a, false, b, (short)0, c, false, false);
        }
    }

    float bv = bias[obase + l];
    #pragma unroll
    for (int r = 0; r < 8; ++r) {
        int Mr = hi ? (r + 8) : r;
        size_t b_ = (size_t)(bbase + Mr);
        out[b_ * (NN * OUTD) + (size_t)n * OUTD + obase + l] = c[r] + bv;
    }
}

extern "C" void kernel_launch(void* const* d_in, const int* in_sizes, int n_in,
                              void* d_out, int out_size, void* d_ws, size_t ws_size,
                              hipStream_t stream) {
    const float* inputs = (const float*)d_in[0];   // [B,N,D]
    const float* adj    = (const float*)d_in[1];   // [E]
    const int*   rows   = (const int*)  d_in[2];   // [E]
    const int*   cols   = (const int*)  d_in[3];   // [E]
    const float* W      = (const float*)d_in[4];   // [320,64]
    const float* bias   = (const float*)d_in[5];   // [64]
    float* out = (float*)d_out;

    float* ws  = (float*)d_ws;
    float* xs0 = ws;                 // 5 planes of [N, 2048]
    float* xs1 = ws + 1 * CH;
    float* xs2 = ws + 2 * CH;
    float* xs3 = ws + 3 * CH;
    float* xs4 = ws + 4 * CH;
    float* drow = ws + 5 * CH;
    float* dcol = drow + NN;
    int* cnt1 = (int*)(dcol + NN);
    int* cnt2 = cnt1 + NN;
    int* off1 = cnt2 + NN;           // NN+1 entries
    int* cur1 = off1 + NN + 1;
    int* off2 = cur1 + NN;           // NN+1 entries
    int* cur2 = off2 + NN + 1;
    int*   src1p = cur2 + NN;        // EE
    float* val1p = (float*)(src1p + EE);
    int*   src2p = (int*)(val1p + EE);
    float* val2p = (float*)(src2p + EE);

    (void)in_sizes; (void)n_in; (void)out_size; (void)ws_size;

    // zero degrees + histograms (contiguous 4*NN words)
    (void)hipMemsetAsync(drow, 0, 4 * NN * sizeof(float), stream);

    // degrees + histograms, scans, bucket fill
    k_deg_cnt<<<EE / 256, 256, 0, stream>>>(adj, rows, cols, drow, dcol, cnt1, cnt2);
    k_scan<<<2, 256, 0, stream>>>(cnt1, off1, cur1, cnt2, off2, cur2);
    k_fill<<<EE / 256, 256, 0, stream>>>(adj, rows, cols, drow, dcol,
                                         cur1, src1p, val1p, cur2, src2p, val2p);

    // x0 pack
    k_pack_x0<<<NN, 256, 0, stream>>>(inputs, xs0);

    // xs1 = S1 @ xs0
    k_spmm_gather<<<NN, 256, 0, stream>>>(xs0, xs0, xs1, off1, src1p, val1p, 1.0f, 0);
    // xs2 = 2*S1 @ xs1 - xs0
    k_spmm_gather<<<NN, 256, 0, stream>>>(xs1, xs0, xs2, off1, src1p, val1p, 2.0f, 1);
    // xs3 = S2 @ xs1   (x0 not reset between supports)
    k_spmm_gather<<<NN, 256, 0, stream>>>(xs1, xs1, xs3, off2, src2p, val2p, 1.0f, 0);
    // xs4 = 2*S2 @ xs3 - xs1
    k_spmm_gather<<<NN, 256, 0, stream>>>(xs3, xs1, xs4, off2, src2p, val2p, 2.0f, 1);

    // out = X @ W + bias via f32 WMMA
    k_gemm_wmma<<<NN, 256, 0, stream>>>(xs0, W, bias, out);
}